// RotaryPositionalEmbedding_41721312313984
// MI455X (gfx1250) — compile-verified
//
#include <hip/hip_runtime.h>
#include <hip/hip_bf16.h>
#include <stdint.h>

// RoPE via block-diagonal rotation-matrix gather.
//   out[2k]   = cos_k*x[2k] - sin_k*x[2k+1]
//   out[2k+1] = sin_k*x[2k] + cos_k*x[2k+1]
// cos_k = R[pos][2k][2k]   -> float offset pos*16384 + 258*k
// sin_k = R[pos][2k+1][2k] -> float offset pos*16384 + 258*k + 128
//
// One wave32 per token. The scattered cos/sin gather goes through the CDNA5
// async VMEM->LDS engine (ASYNCcnt) while the wave's coalesced float4 load of
// x proceeds on LOADcnt in parallel; a single s_wait_asynccnt 0 joins them.
// Each wave reads only the LDS it gathered itself => no workgroup barrier.

#define DK          128
#define PAIRS       64          // DK/2
#define TOK_PER_BLK 8           // waves per block
#define THREADS     256         // 8 * wave32

typedef __attribute__((address_space(1))) int as1_int;   // global
typedef __attribute__((address_space(3))) int as3_int;   // LDS

__device__ __forceinline__ void async_gather_b32(const float* gsrc, float* lds_dst) {
#if __has_builtin(__builtin_amdgcn_global_load_async_to_lds_b32)
    __builtin_amdgcn_global_load_async_to_lds_b32(
        (as1_int*)(uintptr_t)gsrc, (as3_int*)lds_dst, /*imm offset*/0, /*cpol*/0);
#else
    uint32_t lds_off = (uint32_t)(uintptr_t)(as3_int*)lds_dst;
    asm volatile("global_load_async_to_lds_b32 %0, %1, off"
                 :: "v"(lds_off), "v"(gsrc) : "memory");
#endif
}

__device__ __forceinline__ void wait_async0() {
#if __has_builtin(__builtin_amdgcn_s_wait_asynccnt)
    __builtin_amdgcn_s_wait_asynccnt(0);
#else
    asm volatile("s_wait_asynccnt 0" ::: "memory");
#endif
    asm volatile("" ::: "memory");   // keep LDS reads below the wait
}

__global__ __launch_bounds__(THREADS)
void rope_gather_async_kernel(const float* __restrict__ x,
                              const int*   __restrict__ token_positions,
                              const float* __restrict__ R,
                              float*       __restrict__ out,
                              int ntok)
{
    __shared__ float cs[TOK_PER_BLK * DK];   // per wave: [0..63]=cos, [64..127]=sin

    const int w     = threadIdx.x >> 5;      // wave in block
    const int l     = threadIdx.x & 31;      // lane in wave
    const int token = blockIdx.x * TOK_PER_BLK + w;
    if (token >= ntok) return;               // wave-uniform exit

    const int    p   = token_positions[token];
    const float* Rb  = R + (size_t)p * (DK * DK);
    float*       csw = cs + w * DK;

    const int k0 = 2 * l;                    // this lane owns pairs k0, k0+1

    // --- async scattered gather of cos/sin into LDS (ASYNCcnt) ---
    async_gather_b32(Rb + (size_t)k0       * 258,      csw + k0);
    async_gather_b32(Rb + (size_t)(k0 + 1) * 258,      csw + k0 + 1);
    async_gather_b32(Rb + (size_t)k0       * 258 + DK, csw + PAIRS + k0);
    async_gather_b32(Rb + (size_t)(k0 + 1) * 258 + DK, csw + PAIRS + k0 + 1);

    // --- overlapped coalesced x load (LOADcnt), 512B per wave ---
    const float4 v = ((const float4*)(x + (size_t)token * DK))[l];

    wait_async0();                           // this wave's gather complete

    const float2 c  = *(const float2*)(csw + k0);          // cos_{k0}, cos_{k0+1}
    const float2 sn = *(const float2*)(csw + PAIRS + k0);  // sin_{k0}, sin_{k0+1}

    float4 o;
    o.x = c.x * v.x - sn.x * v.y;
    o.y = sn.x * v.x + c.x * v.y;
    o.z = c.y * v.z - sn.y * v.w;
    o.w = sn.y * v.z + c.y * v.w;

    ((float4*)(out + (size_t)token * DK))[l] = o;
}

extern "C" void kernel_launch(void* const* d_in, const int* in_sizes, int n_in,
                              void* d_out, int out_size, void* d_ws, size_t ws_size,
                              hipStream_t stream) {
    // setup_inputs order: x (f32, B*S*128), token_positions (i32, B*S),
    //                     rotation_matrix (f32, 4096*128*128)
    const float* x  = (const float*)d_in[0];
    const int*   tp = (const int*)d_in[1];
    const float* R  = (const float*)d_in[2];
    float*       o  = (float*)d_out;

    const int ntok   = in_sizes[1];                        // B*S tokens
    const int blocks = (ntok + TOK_PER_BLK - 1) / TOK_PER_BLK;

    rope_gather_async_kernel<<<blocks, THREADS, 0, stream>>>(x, tp, R, o, ntok);
}